// HyperbolicMemoryRetrieval_3384434229327
// MI455X (gfx1250) — compile-verified
//
#include <hip/hip_runtime.h>
#include <math.h>

#define EPSF      1e-5f
#define MAXNORM   (1.0f - 1e-5f)
#define MAXN2     (MAXNORM * MAXNORM)
#define TAUF      0.1f
#define ALPHAF    0.1f

#define BB    32
#define SS    2048
#define HH    1024
#define MM    128
#define NMEM  500000
#define NTILE (NMEM / 16)     // 31250
#define NBLKA 256
#define NWAVE (NBLKA * 8)     // 2048 waves in distance pass

typedef __attribute__((ext_vector_type(2))) float v2f;
typedef __attribute__((ext_vector_type(8))) float v8f;

// ---------------------------------------------------------------- helpers
__device__ __forceinline__ float blk_reduce_sum(float v, float* sm, int n) {
  int t = threadIdx.x;
  sm[t] = v; __syncthreads();
  for (int off = n >> 1; off > 0; off >>= 1) {
    if (t < off) sm[t] += sm[t + off];
    __syncthreads();
  }
  float r = sm[0]; __syncthreads();
  return r;
}

__device__ __forceinline__ void topk_insert(float v, int id,
                                            float (&bv)[16], int (&bi)[16]) {
  if (v < bv[15]) {
    bv[15] = v; bi[15] = id;
#pragma unroll
    for (int j = 15; j > 0; j--) {
      if (bv[j] < bv[j - 1]) {
        float tv = bv[j]; bv[j] = bv[j - 1]; bv[j - 1] = tv;
        int   ti = bi[j]; bi[j] = bi[j - 1]; bi[j - 1] = ti;
      }
    }
  }
}

// ------------------------------------------- 1a: partial mean-pool (float4)
__global__ __launch_bounds__(256) void pool_partial_kernel(
    const float4* __restrict__ hs4, float4* __restrict__ partial4) {
  int blk = blockIdx.x;            // 32 b * 8 schunk = 256 blocks
  int b = blk >> 3, sc = blk & 7;
  int t = threadIdx.x;             // one float4 column (h/4) per thread
  const float4* p = hs4 + ((size_t)b * SS + sc * 256) * 256 + t;
  float4 s = make_float4(0.f, 0.f, 0.f, 0.f);
#pragma unroll 8
  for (int i = 0; i < 256; i++) {
    float4 v = p[(size_t)i * 256];
    s.x += v.x; s.y += v.y; s.z += v.z; s.w += v.w;
  }
  partial4[(b * 8 + sc) * 256 + t] = s;
}

// ---------------------------------------------------- 1b: finalize pooling
__global__ __launch_bounds__(256) void pool_final_kernel(
    const float4* __restrict__ partial4, float4* __restrict__ pooled4) {
  int b = blockIdx.x;              // 32 blocks
  int t = threadIdx.x;
  float4 s = make_float4(0.f, 0.f, 0.f, 0.f);
#pragma unroll
  for (int sc = 0; sc < 8; sc++) {
    float4 v = partial4[(b * 8 + sc) * 256 + t];
    s.x += v.x; s.y += v.y; s.z += v.z; s.w += v.w;
  }
  const float inv = 1.0f / (float)SS;
  s.x *= inv; s.y *= inv; s.z *= inv; s.w *= inv;
  pooled4[b * 256 + t] = s;
}

// ------------------------------------- 2: query network + exp-map + project
__global__ __launch_bounds__(128) void query_kernel(
    const float* __restrict__ pooled, const float* __restrict__ W1,
    const float* __restrict__ b1, const float* __restrict__ lng,
    const float* __restrict__ lnb, const float* __restrict__ W2,
    const float* __restrict__ b2, const float* __restrict__ qorigin,
    float* __restrict__ qout, float* __restrict__ qn_raw,
    float* __restrict__ qn_clip) {
  __shared__ float red[128];
  __shared__ float gbuf[128];
  int b = blockIdx.x, m = threadIdx.x;
  const float* p = pooled + b * HH;

  // h = pooled @ W1 + b1
  float acc = b1[m];
  for (int k = 0; k < HH; k++) acc += p[k] * W1[k * MM + m];

  // layernorm
  float mu  = blk_reduce_sum(acc, red, 128) * (1.0f / 128.0f);
  float d   = acc - mu;
  float var = blk_reduce_sum(d * d, red, 128) * (1.0f / 128.0f);
  float xn  = d * rsqrtf(var + 1e-5f) * lng[m] + lnb[m];

  // exact gelu
  float g = 0.5f * xn * (1.0f + erff(xn * 0.70710678118654752f));
  gbuf[m] = g; __syncthreads();

  // tangent = gelu(ln(h)) @ W2 + b2
  float tg = b2[m];
  for (int j = 0; j < MM; j++) tg += gbuf[j] * W2[j * MM + m];
  __syncthreads();

  // origin = project(q_origin)
  float o   = qorigin[m];
  float on2 = blk_reduce_sum(o * o, red, 128);
  float onn = fmaxf(sqrtf(on2), EPSF);
  o = o / fmaxf(onn / MAXNORM, 1.0f);

  // second = tanh(|t|/2) * t / |t|
  float tn2 = blk_reduce_sum(tg * tg, red, 128);
  float vn  = fmaxf(sqrtf(tn2), EPSF);
  float sec = tanhf(0.5f * vn) / vn * tg;

  // mobius_add(o, sec)
  float nu  = blk_reduce_sum(o * o, red, 128);
  nu = fminf(fmaxf(nu, 0.f), MAXN2);
  float nv  = blk_reduce_sum(sec * sec, red, 128);
  nv = fminf(fmaxf(nv, 0.f), MAXN2);
  float duv = blk_reduce_sum(o * sec, red, 128);
  float num = (1.0f + 2.0f * duv + nv) * o + (1.0f - nu) * sec;
  float den = fmaxf(1.0f + 2.0f * duv + nu * nv, EPSF);
  float r   = num / den;

  // project
  float rn2 = blk_reduce_sum(r * r, red, 128);
  float rn  = fmaxf(sqrtf(rn2), EPSF);
  float f   = fmaxf(rn / MAXNORM, 1.0f);
  float q   = r / f;
  qout[b * MM + m] = q;
  if (m == 0) {
    float qraw = rn2 / (f * f);
    qn_raw[b]  = qraw;
    qn_clip[b] = fminf(fmaxf(qraw, 0.f), MAXN2);
  }
}

// --------------------- 3: WMMA gram + hyperbolic distance proxy + top-k
__global__ __launch_bounds__(256, 1) void dist_topk_kernel(
    const float* __restrict__ memv, const float* __restrict__ qv,
    const float* __restrict__ qn_raw, const float* __restrict__ qn_clip,
    float* __restrict__ cval, int* __restrict__ cidx) {
  __shared__ float qs[32 * 132];           // padded: stride 132 avoids conflicts
  int tid = threadIdx.x;
  for (int i = tid; i < 32 * 128; i += 256) {
    int b = i >> 7, c = i & 127;
    qs[b * 132 + c] = qv[i];
  }
  __syncthreads();

  int lane  = tid & 31;
  int wave  = tid >> 5;
  int gwave = blockIdx.x * 8 + wave;

  int bq = lane & 15;                      // output column (batch) of this lane
  int hi = lane >> 4;                      // lane half (K pair / M offset select)

  float qr0 = qn_raw[bq],       qr1 = qn_raw[bq + 16];
  float i0  = 1.0f - qn_clip[bq], i1 = 1.0f - qn_clip[bq + 16];

  float bv0[16], bv1[16]; int bi0[16], bi1[16];
#pragma unroll
  for (int k = 0; k < 16; k++) {
    bv0[k] = __builtin_inff(); bv1[k] = __builtin_inff();
    bi0[k] = 0; bi1[k] = 0;
  }

  for (int tile = gwave; tile < NTILE; tile += NWAVE) {
    const float* arow  = memv + (size_t)(tile * 16 + bq) * MM + 2 * hi;
    const float* brow0 = qs + bq * 132 + 2 * hi;
    const float* brow1 = qs + (bq + 16) * 132 + 2 * hi;

    // speculative prefetch of next tile for this wave
    {
      int nt = tile + NWAVE; if (nt >= NTILE) nt = tile;
      __builtin_prefetch(memv + (size_t)(nt * 16 + bq) * MM + 2 * hi, 0, 1);
    }

    v8f acc0 = {}; v8f acc1 = {};
    float sq = 0.f;

    // 4-deep software pipeline on A fragments: keep 4 loads in flight
    v2f abuf[4];
#pragma unroll
    for (int s = 0; s < 4; s++) abuf[s] = *(const v2f*)(arow + 4 * s);

#pragma unroll
    for (int s = 0; s < 32; s++) {
      v2f a = abuf[s & 3];
      if (s + 4 < 32) abuf[s & 3] = *(const v2f*)(arow + 4 * (s + 4));
      v2f b0 = *(const v2f*)(brow0 + 4 * s);
      v2f b1 = *(const v2f*)(brow1 + 4 * s);
      sq += a.x * a.x + a.y * a.y;
      acc0 = __builtin_amdgcn_wmma_f32_16x16x4_f32(
          false, a, false, b0, (short)0, acc0, false, false);
      acc1 = __builtin_amdgcn_wmma_f32_16x16x4_f32(
          false, a, false, b1, (short)0, acc1, false, false);
    }
    // complete ||mem_row||^2 : other lane-half holds the other K residues
    sq += __shfl_xor(sq, 16, 32);          // lane j now holds mn of row (j&15)

#pragma unroll
    for (int r = 0; r < 8; r++) {
      int   mrow = r + (hi ? 8 : 0);       // C/D layout: lanes16-31 -> M += 8
      float mnr  = __shfl(sq, mrow, 32);
      float mnc  = fminf(fmaxf(mnr, 0.f), MAXN2);
      float invm = 1.0f - mnc;
      int   gidx = tile * 16 + mrow;
      {
        float g   = acc0[r];
        float ds  = fmaxf(qr0 + mnr - 2.0f * g, 0.f);
        float dn  = fmaxf(i0 * invm, EPSF);
        topk_insert(2.0f * ds / dn, gidx, bv0, bi0);
      }
      {
        float g   = acc1[r];
        float ds  = fmaxf(qr1 + mnr - 2.0f * g, 0.f);
        float dn  = fmaxf(i1 * invm, EPSF);
        topk_insert(2.0f * ds / dn, gidx, bv1, bi1);
      }
    }
  }

  int base0 = ((bq * NWAVE + gwave) * 2 + hi) * 16;
  int base1 = (((bq + 16) * NWAVE + gwave) * 2 + hi) * 16;
#pragma unroll
  for (int k = 0; k < 16; k++) {
    cval[base0 + k] = bv0[k]; cidx[base0 + k] = bi0[k];
    cval[base1 + k] = bv1[k]; cidx[base1 + k] = bi1[k];
  }
}

// --------------------- 4: exact top-16 merge, softmax, gather, projection
__global__ __launch_bounds__(256, 1) void merge_retrieve_kernel(
    const float* __restrict__ cval, const int* __restrict__ cidx,
    const float* __restrict__ memv, const float* __restrict__ Wp,
    const float* __restrict__ bp, float* __restrict__ force) {
  __shared__ float hv[256]; __shared__ int hid[256]; __shared__ int hth[256];
  __shared__ float selv[16]; __shared__ int seli[16];
  __shared__ float wk[16];
  __shared__ float rbuf[128];

  int b = blockIdx.x, t = threadIdx.x;
  const float* cv = cval + (size_t)b * NWAVE * 32;   // 65536 candidates
  const int*   ci = cidx + (size_t)b * NWAVE * 32;

  float lv[16]; int li[16];
#pragma unroll
  for (int k = 0; k < 16; k++) { lv[k] = __builtin_inff(); li[k] = 0x7fffffff; }
  for (int j = 0; j < (NWAVE * 32) / 256; j++) {
    int p = j * 256 + t;                   // coalesced
    topk_insert(cv[p], ci[p], lv, li);
  }

  for (int k = 0; k < 16; k++) {
    hv[t] = lv[0]; hid[t] = li[0]; hth[t] = t;
    __syncthreads();
    for (int off = 128; off > 0; off >>= 1) {
      if (t < off) {
        float v2 = hv[t + off]; int i2 = hid[t + off];
        if (v2 < hv[t] || (v2 == hv[t] && i2 < hid[t])) {
          hv[t] = v2; hid[t] = i2; hth[t] = hth[t + off];
        }
      }
      __syncthreads();
    }
    if (t == 0) { selv[k] = hv[0]; seli[k] = hid[0]; }
    int w = hth[0];
    __syncthreads();
    if (t == w) {                          // winner pops its head
#pragma unroll
      for (int j = 0; j < 15; j++) { lv[j] = lv[j + 1]; li[j] = li[j + 1]; }
      lv[15] = __builtin_inff();
    }
    __syncthreads();
  }

  if (t == 0) {                            // softmax(-arccosh(1+x)/tau)
    float dd[16]; float mx = -__builtin_inff();
#pragma unroll
    for (int k = 0; k < 16; k++) {
      dd[k] = -acoshf(1.0f + selv[k]) / TAUF;
      mx = fmaxf(mx, dd[k]);
    }
    float ssum = 0.f;
#pragma unroll
    for (int k = 0; k < 16; k++) { dd[k] = expf(dd[k] - mx); ssum += dd[k]; }
#pragma unroll
    for (int k = 0; k < 16; k++) wk[k] = dd[k] / ssum;
  }
  __syncthreads();

  if (t < 128) {                           // retrieved = sum_k w_k * mem[idx_k]
    float r = 0.f;
#pragma unroll
    for (int k = 0; k < 16; k++) r += wk[k] * memv[(size_t)seli[k] * MM + t];
    rbuf[t] = r;
  }
  __syncthreads();

  for (int hh = t; hh < HH; hh += 256) {   // force = retrieved @ Wp + bp
    float f = bp[hh];
    for (int m = 0; m < MM; m++) f += rbuf[m] * Wp[m * HH + hh];
    force[b * HH + hh] = f;
  }
}

// --------------------------------- 5: out = hidden + alpha * force (bcast)
__global__ __launch_bounds__(256) void add_force_kernel(
    const float4* __restrict__ hs, const float4* __restrict__ force4,
    float4* __restrict__ out) {
  unsigned i = blockIdx.x * 256u + threadIdx.x;   // 16,777,216 float4s
  unsigned b = i >> 19;                           // 2048*1024/4 per batch
  unsigned h = i & 255u;                          // 1024/4 per row
  float4 f = force4[b * 256u + h];
  float4 v = hs[i];
  v.x += ALPHAF * f.x; v.y += ALPHAF * f.y;
  v.z += ALPHAF * f.z; v.w += ALPHAF * f.w;
  out[i] = v;
}

// ------------------------------------------------------------------ launch
extern "C" void kernel_launch(void* const* d_in, const int* in_sizes, int n_in,
                              void* d_out, int out_size, void* d_ws,
                              size_t ws_size, hipStream_t stream) {
  (void)in_sizes; (void)n_in; (void)out_size; (void)ws_size;
  const float* hs      = (const float*)d_in[0];
  const float* W1      = (const float*)d_in[1];
  const float* b1      = (const float*)d_in[2];
  const float* lng     = (const float*)d_in[3];
  const float* lnb     = (const float*)d_in[4];
  const float* W2      = (const float*)d_in[5];
  const float* b2      = (const float*)d_in[6];
  const float* qorigin = (const float*)d_in[7];
  const float* memv    = (const float*)d_in[8];
  const float* Wp      = (const float*)d_in[9];
  const float* bp      = (const float*)d_in[10];
  float* out = (float*)d_out;
  char*  ws  = (char*)d_ws;

  float* partial = (float*)(ws + 0);            // 32*8*1024 f   = 1.00 MB
  float* pooled  = (float*)(ws + 1048576);      // 32*1024   f   = 128 KB
  float* qv      = (float*)(ws + 1179648);      // 32*128    f   = 16 KB
  float* qn_raw  = (float*)(ws + 1196032);      // 32 f
  float* qn_clip = (float*)(ws + 1196160);      // 32 f
  float* cval    = (float*)(ws + 1196288);      // 32*2048*32 f  = 8 MB
  int*   cidx    = (int*)  (ws + 9584896);      // 32*2048*32 i  = 8 MB
  float* force   = (float*)(ws + 17973504);     // 32*1024   f   = 128 KB

  pool_partial_kernel<<<dim3(256), dim3(256), 0, stream>>>(
      (const float4*)hs, (float4*)partial);
  pool_final_kernel<<<dim3(BB), dim3(256), 0, stream>>>(
      (const float4*)partial, (float4*)pooled);
  query_kernel<<<dim3(BB), dim3(128), 0, stream>>>(
      pooled, W1, b1, lng, lnb, W2, b2, qorigin, qv, qn_raw, qn_clip);
  dist_topk_kernel<<<dim3(NBLKA), dim3(256), 0, stream>>>(
      memv, qv, qn_raw, qn_clip, cval, cidx);
  merge_retrieve_kernel<<<dim3(BB), dim3(256), 0, stream>>>(
      cval, cidx, memv, Wp, bp, force);
  add_force_kernel<<<dim3(65536), dim3(256), 0, stream>>>(
      (const float4*)hs, (const float4*)force, (float4*)out);
}